// ImageStitchingLayer_24790551232735
// MI455X (gfx1250) — compile-verified
//
#include <hip/hip_runtime.h>
#include <hip/hip_bf16.h>

// ---------------------------------------------------------------------------
// CDNA5 async global->LDS streaming helpers (ASYNCcnt path, gfx1250)
// ---------------------------------------------------------------------------
__device__ __forceinline__ void async_ld_b128(unsigned ldsOff, unsigned gOff,
                                              const float* base) {
    // GVS mode: saddr(64-bit SGPR base) + vaddr(32-bit per-lane offset)
    asm volatile("global_load_async_to_lds_b128 %0, %1, %2"
                 :
                 : "v"(ldsOff), "v"(gOff), "s"(base)
                 : "memory");
}

__device__ __forceinline__ void wait_async_pairs(int rem_pairs) {
    // rem_pairs = pairs of B128 ops that may remain outstanding AFTER the wait
    switch (rem_pairs) {
        case 0:  asm volatile("s_wait_asynccnt 0x0" ::: "memory"); break;
        case 1:  asm volatile("s_wait_asynccnt 0x2" ::: "memory"); break;
        case 2:  asm volatile("s_wait_asynccnt 0x4" ::: "memory"); break;
        default: asm volatile("s_wait_asynccnt 0x6" ::: "memory"); break;
    }
}

__device__ __forceinline__ void wait_ds_done() {
    // Guarantee all prior ds_loads have returned data before we let the async
    // engine overwrite their source buffer (async LDS writes are unordered
    // with the DS pipe).
    asm volatile("s_wait_dscnt 0x0" ::: "memory");
}

// ---------------------------------------------------------------------------
// Stage 1: per-block partial statistics for every (pair, channel)
// grid = 1024 blocks: [pair:4][zchunk:4][colgroup:64], 256 threads
// ---------------------------------------------------------------------------
__global__ __launch_bounds__(256) void stitch_partials(
        const float* __restrict__ stacks,   // (4, 48, 512, 512, 2) f32
        const float* __restrict__ zw,       // (4,) f32
        float* __restrict__ ws) {           // (1024, 10) f32 partials
    __shared__ float4 stA[4][256];          // 16 KB, 4-deep pipeline, stack i
    __shared__ float4 stB[4][256];          // 16 KB, stack j
    __shared__ float  red[8][10];

    const int tid  = threadIdx.x;
    const int pair = blockIdx.x >> 8;
    const int sub  = blockIdx.x & 255;
    const int zc   = sub >> 6;              // z chunk 0..3
    const int cg   = sub & 63;              // column group 0..63

    const int It[4] = {1, 2, 3, 3};
    const int Jt[4] = {0, 0, 1, 2};
    const int Vt[4] = {0, 1, 1, 0};         // 1 = vertical overlap
    const int i = It[pair], j = Jt[pair], vert = Vt[pair];

    const float wi = zw[i]; const float fi = wi - floorf(wi); const float gi = 1.0f - fi;
    const float wj = zw[j]; const float fj = wj - floorf(wj); const float gj = 1.0f - fj;

    // Region layout per z-plane: rows of contiguous floats.
    // horz: 512 rows of 128 floats (w in [0,64) x 2ch), offsets: o1 w=0, o2 w=448
    // vert: 64 rows of 1024 floats (w in [0,512) x 2ch), offsets: o1 h=0, o2 h=448
    const int col      = (cg * 256 + tid) * 4;          // float index in region plane
    const int rowshift = vert ? 10 : 7;
    const int h        = col >> rowshift;
    const int roff     = col & ((1 << rowshift) - 1);   // even -> x,z ch0; y,w ch1
    const int h2off    = vert ? 448 : 0;
    const int w2off    = vert ? 0 : 448;

    // z output range [lo,hi) of 49; chunk 3 also covers zo=48 (pure tail)
    const int lo = zc * 12;
    const int hi = (zc == 3) ? 49 : lo + 12;
    const int zstart = lo ? lo - 1 : 0;                 // extra plane for prev
    const int zend   = (hi < 48) ? hi : 48;
    const int P      = zend - zstart;                   // 12 or 13 planes

    const unsigned SB = 48u * 512u * 512u * 2u;         // floats per stack
    const unsigned ZS = 512u * 512u * 2u * 4u;          // z-plane stride bytes
    unsigned g1 = ((unsigned)i * SB +
                   (unsigned)((zstart * 512 + h) * 1024 + roff)) * 4u;
    unsigned g2 = ((unsigned)j * SB +
                   (unsigned)((zstart * 512 + h + h2off) * 1024 + w2off * 2 + roff)) * 4u;

    // pipeline prologue: 3 planes in flight (6 async B128 ops / lane)
#pragma unroll
    for (int t = 0; t < 3; ++t) {
        async_ld_b128((unsigned)(size_t)&stA[t][tid], g1 + (unsigned)t * ZS, stacks);
        async_ld_b128((unsigned)(size_t)&stB[t][tid], g2 + (unsigned)t * ZS, stacks);
    }

    float S1c0=0.f,S1c1=0.f,S2c0=0.f,S2c1=0.f,Q1c0=0.f,Q1c1=0.f,
          Q2c0=0.f,Q2c1=0.f,Pc0=0.f,Pc1=0.f;
    float4 pva = make_float4(0.f,0.f,0.f,0.f);
    float4 pvb = make_float4(0.f,0.f,0.f,0.f);

    auto step = [&](const float4& a, const float4& b) {
        const float o1x = gi*a.x + fi*pva.x, o1y = gi*a.y + fi*pva.y;
        const float o1z = gi*a.z + fi*pva.z, o1w = gi*a.w + fi*pva.w;
        const float o2x = gj*b.x + fj*pvb.x, o2y = gj*b.y + fj*pvb.y;
        const float o2z = gj*b.z + fj*pvb.z, o2w = gj*b.w + fj*pvb.w;
        S1c0 += o1x + o1z;  S1c1 += o1y + o1w;
        S2c0 += o2x + o2z;  S2c1 += o2y + o2w;
        Q1c0 = fmaf(o1x,o1x,Q1c0); Q1c0 = fmaf(o1z,o1z,Q1c0);
        Q1c1 = fmaf(o1y,o1y,Q1c1); Q1c1 = fmaf(o1w,o1w,Q1c1);
        Q2c0 = fmaf(o2x,o2x,Q2c0); Q2c0 = fmaf(o2z,o2z,Q2c0);
        Q2c1 = fmaf(o2y,o2y,Q2c1); Q2c1 = fmaf(o2w,o2w,Q2c1);
        Pc0  = fmaf(o1x,o2x,Pc0);  Pc0  = fmaf(o1z,o2z,Pc0);
        Pc1  = fmaf(o1y,o2y,Pc1);  Pc1  = fmaf(o1w,o2w,Pc1);
        pva = a; pvb = b;
    };

    for (int t = 0; t < P; ++t) {
        if (t + 3 < P) {
            // Buffer (t+3)&3 == (t-1)&3: make sure last iteration's ds_loads
            // have fully returned before the async engine may overwrite it.
            wait_ds_done();
            const unsigned o = (unsigned)(t + 3) * ZS;
            const int bf = (t + 3) & 3;
            async_ld_b128((unsigned)(size_t)&stA[bf][tid], g1 + o, stacks);
            async_ld_b128((unsigned)(size_t)&stB[bf][tid], g2 + o, stacks);
            wait_async_pairs(3);          // plane t guaranteed landed
        } else {
            wait_async_pairs(P - 1 - t);  // drain tail
        }
        const float4 a  = stA[t & 3][tid];
        const float4 b4 = stB[t & 3][tid];
        if (t == 0 && lo > 0) { pva = a; pvb = b4; }  // prev-only plane
        else                  { step(a, b4); }
    }
    if (hi == 49) {                        // zo = 48: cur plane is zero
        const float4 zz = make_float4(0.f,0.f,0.f,0.f);
        step(zz, zz);
    }

    // wave32 xor reduction, then cross-wave via LDS
    float acc[10] = {S1c0,S1c1,S2c0,S2c1,Q1c0,Q1c1,Q2c0,Q2c1,Pc0,Pc1};
#pragma unroll
    for (int k = 0; k < 10; ++k) {
        float v = acc[k];
#pragma unroll
        for (int m = 16; m >= 1; m >>= 1) v += __shfl_xor(v, m, 32);
        acc[k] = v;
    }
    const int wv = tid >> 5;
    if ((tid & 31) == 0) {
#pragma unroll
        for (int k = 0; k < 10; ++k) red[wv][k] = acc[k];
    }
    __syncthreads();
    if (tid < 10) {
        float t0 = 0.f;
#pragma unroll
        for (int w = 0; w < 8; ++w) t0 += red[w][tid];
        ws[(unsigned)blockIdx.x * 10u + (unsigned)tid] = t0;
    }
}

// ---------------------------------------------------------------------------
// Stage 2: deterministic final reduction + closed-form loss
// ---------------------------------------------------------------------------
__global__ __launch_bounds__(64) void stitch_final(const float* __restrict__ ws,
                                                   float* __restrict__ out) {
    __shared__ float tot[40];
    const int t = threadIdx.x;
    if (t < 40) {
        const int p = t / 10, s = t % 10;
        float acc = 0.f;
        for (int b = 0; b < 256; ++b) acc += ws[(unsigned)(((p << 8) | b) * 10 + s)];
        tot[t] = acc;
    }
    __syncthreads();
    if (t == 0) {
        const double N = 49.0 * 64.0 * 512.0;
        double loss = 0.0;
        for (int p = 0; p < 4; ++p) {
            const float* T = &tot[p * 10];
            for (int c = 0; c < 2; ++c) {
                const double s1 = T[0 + c], s2 = T[2 + c];
                const double q1 = T[4 + c], q2 = T[6 + c], pr = T[8 + c];
                const double m12 = pr - s1 * s2 / N;
                const double m11 = q1 - s1 * s1 / N;
                const double m22 = q2 - s2 * s2 / N;
                loss += m12 * m12 + m11 * m22;
            }
        }
        out[0] = (float)loss;
    }
}

extern "C" void kernel_launch(void* const* d_in, const int* in_sizes, int n_in,
                              void* d_out, int out_size, void* d_ws, size_t ws_size,
                              hipStream_t stream) {
    (void)in_sizes; (void)n_in; (void)out_size; (void)ws_size;
    const float* stacks = (const float*)d_in[0];
    const float* zweights = (const float*)d_in[1];
    float* ws = (float*)d_ws;               // needs 1024*10*4 = 40 KB
    stitch_partials<<<dim3(1024), dim3(256), 0, stream>>>(stacks, zweights, ws);
    stitch_final<<<dim3(1), dim3(64), 0, stream>>>(ws, (float*)d_out);
}